// H2GCN_5342939316785
// MI455X (gfx1250) — compile-verified
//
#include <hip/hip_runtime.h>

#define DIM  128
#define OUTD 16

typedef __attribute__((ext_vector_type(2))) float v2f;
typedef __attribute__((ext_vector_type(4))) float v4f;
typedef __attribute__((ext_vector_type(8))) float v8f;

// Address-space(1) = global. Forcing loads onto the GLOBAL path (LOADcnt only)
// instead of FLAT (LOADcnt + DScnt + LDS arbitration).
typedef __attribute__((address_space(1))) const float*  gcf;
typedef __attribute__((address_space(1))) const v4f*    gcf4;
typedef __attribute__((address_space(1))) const int*    gci;
typedef __attribute__((address_space(1))) float*        gf;
#define TO_GC(p)  ((gcf)(unsigned long long)(p))
#define TO_GC4(p) ((gcf4)(unsigned long long)(p))
#define TO_GCI(p) ((gci)(unsigned long long)(p))
#define TO_G(p)   ((gf)(unsigned long long)(p))

// Non-returning f32 global atomic add (single hardware op, STOREcnt-tracked,
// drained by s_endpgm's implicit wait-idle).
__device__ __forceinline__ void gatomic_add_f32(float* p, float v) {
  asm volatile("global_atomic_add_f32 %0, %1, off" :: "v"(p), "v"(v) : "memory");
}

// ---------------- workspace zeroing ----------------
__global__ void zero_f32(float* __restrict__ p, long long n) {
  gf g = TO_G(p);
  long long i = (long long)blockIdx.x * blockDim.x + threadIdx.x;
  long long stride = (long long)gridDim.x * blockDim.x;
  for (; i < n; i += stride) g[i] = 0.0f;
}

// ---------------- degree ----------------
__global__ void deg_count(const int* __restrict__ rows, float* __restrict__ deg, int ne) {
  int e = blockIdx.x * blockDim.x + threadIdx.x;
  if (e < ne) gatomic_add_f32(&deg[TO_GCI(rows)[e]], 1.0f);
}

__global__ void deg_inv_k(const float* __restrict__ deg, float* __restrict__ dinv, int n) {
  int i = blockIdx.x * blockDim.x + threadIdx.x;
  if (i < n) TO_G(dinv)[i] = 1.0f / fmaxf(TO_GC(deg)[i], 1.0f);
}

// ---------------- SpMM: dst[row] += src[col], one wave per edge ----------------
__global__ void spmm_atomic(const int* __restrict__ rows, const int* __restrict__ cols,
                            const float* __restrict__ src, float* __restrict__ dst, int ne) {
  int gw = (int)((blockIdx.x * (long long)blockDim.x + threadIdx.x) >> 5);
  int lane = threadIdx.x & 31;
  if (gw >= ne) return;
  int r = TO_GCI(rows)[gw];
  int c = TO_GCI(cols)[gw];
  v4f v = TO_GC4(src + (long long)c * DIM)[lane];      // global_load_b128, L2-resident
  float* d = dst + (long long)r * DIM + lane * 4;
  gatomic_add_f32(d + 0, v.x);
  gatomic_add_f32(d + 1, v.y);
  gatomic_add_f32(d + 2, v.z);
  gatomic_add_f32(d + 3, v.w);
}

// ---------------- fused dense path: one wave per 16-node tile ----------------
// stage 1: h_b = relu(scale_b * X_b[16x128] @ W_b[128x128] + b_b)   (b = ego,1hop,2hop)
// stage 2: out[16x16] += h_b @ W_cls[b*128 : (b+1)*128, :]
__global__ __launch_bounds__(128) void h2gcn_fused(
    const float* __restrict__ x,  const float* __restrict__ ax, const float* __restrict__ aax,
    const float* __restrict__ dinv,
    const float* __restrict__ W0, const float* __restrict__ b0,
    const float* __restrict__ W1, const float* __restrict__ b1,
    const float* __restrict__ W2, const float* __restrict__ b2,
    const float* __restrict__ Wc, const float* __restrict__ bc,
    float* __restrict__ out, int n_tiles)
{
  __shared__ float hbuf[4][16 * DIM];          // one 16x128 h-tile per wave (32 KB)
  const int lane = threadIdx.x & 31;
  const int wib  = threadIdx.x >> 5;
  const int n    = lane & 15;                  // N index (and A-matrix row M for this lane)
  const int kh   = lane >> 4;                  // lane half: K pair select
  float* __restrict__ ht = hbuf[wib];

  const gcf Xs[3] = {TO_GC(x),  TO_GC(ax), TO_GC(aax)};
  const gcf Ws[3] = {TO_GC(W0), TO_GC(W1), TO_GC(W2)};
  const gcf Bb[3] = {TO_GC(b0), TO_GC(b1), TO_GC(b2)};
  const gcf Wcg   = TO_GC(Wc);

  const int gwave  = blockIdx.x * 4 + wib;
  const int nwaves = gridDim.x * 4;

  for (int tile = gwave; tile < n_tiles; tile += nwaves) {
    const int   row_a = tile * 16 + n;         // this lane's A-matrix row
    const float di    = TO_GC(dinv)[row_a];
    const float scale[3] = {1.0f, di, di * di};

    v8f oacc = {};                             // 16x16 output fragment (b_cls added at end)

    for (int br = 0; br < 3; ++br) {
      const gcf  X = Xs[br];
      const gcf  W = Ws[br];
      const float s = scale[br];

      v8f hacc[8] = {};                        // eight 16x16 column tiles of h

      const gcf xrow = X + (long long)row_a * DIM;
      for (int k0 = 0; k0 < DIM; k0 += 4) {
        const int k = k0 + 2 * kh;
        v2f a;                                 // A frag: lane row=n, K={k,k+1}
        a.x = xrow[k] * s;                     // merged global_load_b64 + v_pk_mul
        a.y = xrow[k + 1] * s;
        #pragma unroll
        for (int t = 0; t < 8; ++t) {
          v2f b;                               // B frag: rows K={k,k+1}, col = t*16+n
          b.x = W[(long long)k       * DIM + t * 16 + n];
          b.y = W[(long long)(k + 1) * DIM + t * 16 + n];
          hacc[t] = __builtin_amdgcn_wmma_f32_16x16x4_f32(
              false, a, false, b, (short)0, hacc[t], false, false);
        }
      }

      // bias + relu, spill to LDS to convert C-layout -> A-layout (transpose staging)
      #pragma unroll
      for (int t = 0; t < 8; ++t) {
        const float bias = Bb[br][t * 16 + n];
        #pragma unroll
        for (int r = 0; r < 8; ++r) {
          float v = hacc[t][r] + bias;
          v = fmaxf(v, 0.0f);
          ht[(r + 8 * kh) * DIM + t * 16 + n] = v;   // h[M][col]
        }
      }
      // same-wave DS ops are in-order; loads below see the stores above

      // stage 2: h[16x128] @ Wc_slice[128x16] accumulated into oacc
      for (int k0 = 0; k0 < DIM; k0 += 4) {
        const int k = k0 + 2 * kh;
        v2f a2, b2;
        a2.x = ht[n * DIM + k];                // ds_load_b64 (contiguous K pair)
        a2.y = ht[n * DIM + k + 1];
        b2.x = Wcg[(long long)(br * DIM + k)     * OUTD + n];
        b2.y = Wcg[(long long)(br * DIM + k + 1) * OUTD + n];
        oacc = __builtin_amdgcn_wmma_f32_16x16x4_f32(
            false, a2, false, b2, (short)0, oacc, false, false);
      }
    }

    const float bcn = TO_GC(bc)[n];
    gf og = TO_G(out);
    #pragma unroll
    for (int r = 0; r < 8; ++r)
      og[(long long)(tile * 16 + r + 8 * kh) * OUTD + n] = oacc[r] + bcn;
  }
}

extern "C" void kernel_launch(void* const* d_in, const int* in_sizes, int n_in,
                              void* d_out, int out_size, void* d_ws, size_t ws_size,
                              hipStream_t stream) {
  const float* x     = (const float*)d_in[0];
  const int*   ei    = (const int*)  d_in[1];   // int32 (JAX default x64-off)
  const float* W_ego = (const float*)d_in[2];
  const float* b_ego = (const float*)d_in[3];
  const float* W_h1  = (const float*)d_in[4];
  const float* b_h1  = (const float*)d_in[5];
  const float* W_h2  = (const float*)d_in[6];
  const float* b_h2  = (const float*)d_in[7];
  const float* W_cls = (const float*)d_in[8];
  const float* b_cls = (const float*)d_in[9];
  float* out = (float*)d_out;

  const int n  = in_sizes[0] / DIM;   // 100000 nodes
  const int ne = in_sizes[1] / 2;     // 1.6M edges

  // workspace layout: [deg n][dinv n][ax n*128][aax n*128]
  float* ws   = (float*)d_ws;
  float* deg  = ws;
  float* dinv = ws + n;
  float* ax_  = ws + 2LL * n;
  float* aax_ = ax_ + (long long)n * DIM;

  // zero deg + dinv + ax + aax
  long long ztot = (long long)n * (2 + 2 * DIM);
  zero_f32<<<2048, 256, 0, stream>>>(ws, ztot);

  const int* rows = ei;
  const int* cols = ei + ne;

  deg_count<<<(ne + 255) / 256, 256, 0, stream>>>(rows, deg, ne);
  deg_inv_k<<<(n + 255) / 256, 256, 0, stream>>>(deg, dinv, n);

  spmm_atomic<<<(ne + 7) / 8, 256, 0, stream>>>(rows, cols, x,   ax_,  ne);  // ax  = A x
  spmm_atomic<<<(ne + 7) / 8, 256, 0, stream>>>(rows, cols, ax_, aax_, ne);  // aax = A(Ax)

  const int n_tiles = n / 16;             // 6250
  const int blocks  = (n_tiles + 3) / 4;  // one tile per wave, 4 waves/block
  h2gcn_fused<<<blocks, 128, 0, stream>>>(x, ax_, aax_, dinv,
                                          W_ego, b_ego, W_h1, b_h1, W_h2, b_h2,
                                          W_cls, b_cls, out, n_tiles);
}